// MSREnhancement_55619826483298
// MI455X (gfx1250) — compile-verified
//
#include <hip/hip_runtime.h>

typedef __attribute__((ext_vector_type(2))) float v2f;
typedef __attribute__((ext_vector_type(8))) float v8f;

#define NPLANES 24          // 8 batches * 3 channels, each a contiguous HxW plane
#define EPSF 1e-6f

// ---------------------------------------------------------------------------
// reflect-101 index fold (iterated, matching jnp.pad('reflect') iterated pads)
// ---------------------------------------------------------------------------
__device__ __forceinline__ int foldidx(int i, int n) {
  while (i < 0 || i > n - 1) {
    if (i < 0) i = -i;
    if (i > n - 1) i = 2 * (n - 1) - i;
  }
  return i;
}

// ---------------------------------------------------------------------------
// pyrDown: 5x5 binomial (1,4,6,4,1)/16 separable, reflect-101, decimate by 2.
// LDS-tiled: 36x36 input tile -> 16x16 output tile. fromInput: v = v*255+1.
// ---------------------------------------------------------------------------
__global__ void pyrdown_kernel(const float* __restrict__ src, float* __restrict__ dst,
                               int n, int fromInput) {
  const int no = n >> 1;
  __shared__ float tile[36 * 36];
  const int p = blockIdx.z;
  const float* sp = src + (size_t)p * n * n;
  float* dp = dst + (size_t)p * no * no;
  const int bx = blockIdx.x * 16, by = blockIdx.y * 16;
  const int iy0 = 2 * by - 2, ix0 = 2 * bx - 2;
  const int tid = threadIdx.y * 16 + threadIdx.x;
  for (int i = tid; i < 36 * 36; i += 256) {
    int r = i / 36, c = i % 36;
    int gy = foldidx(iy0 + r, n);
    int gx = foldidx(ix0 + c, n);
    float v = sp[gy * n + gx];
    if (fromInput) v = v * 255.0f + 1.0f;
    tile[i] = v;
  }
  __syncthreads();
  const int ox = bx + threadIdx.x, oy = by + threadIdx.y;
  if (ox < no && oy < no) {
    const float kw[5] = {1.f, 4.f, 6.f, 4.f, 1.f};
    float acc = 0.f;
#pragma unroll
    for (int i = 0; i < 5; ++i) {
      float rsum = 0.f;
#pragma unroll
      for (int j = 0; j < 5; ++j)
        rsum += kw[j] * tile[(2 * threadIdx.y + i) * 36 + (2 * threadIdx.x + j)];
      acc += kw[i] * rsum;
    }
    dp[oy * no + ox] = acc * (1.0f / 256.0f);
  }
}

// ---------------------------------------------------------------------------
// Build dense n x n Gaussian blur matrix A (zero-padded to npad x npad):
//   out_row = in_row * A  performs the 1D conv with reflect-101 borders.
//   A[fold(j+t-r)][j] += k[t]
// cv2 sigma-from-ksize: sigma = 0.3*((ksize-1)*0.5 - 1) + 0.8
// ---------------------------------------------------------------------------
__global__ void build_blur_matrix_kernel(float* __restrict__ A, int n, int npad, int ksize) {
  if (threadIdx.x != 0 || blockIdx.x != 0) return;
  float k[9];
  const int r = ksize / 2;
  const float sigma = 0.3f * ((ksize - 1) * 0.5f - 1.0f) + 0.8f;
  float s = 0.f;
  for (int t = 0; t < ksize; ++t) {
    float xv = (float)t - (ksize - 1) * 0.5f;
    k[t] = expf(-xv * xv / (2.f * sigma * sigma));
    s += k[t];
  }
  for (int t = 0; t < ksize; ++t) k[t] /= s;
  for (int i = 0; i < npad * npad; ++i) A[i] = 0.f;
  for (int j = 0; j < n; ++j)
    for (int t = 0; t < ksize; ++t) {
      int pidx = foldidx(j + t - r, n);
      A[pidx * npad + j] += k[t];
    }
}

// ---------------------------------------------------------------------------
// Separable Gaussian blur as dense matmul on small pyramid levels:
//   Out = A^T * (In * A), done with V_WMMA_F32_16X16X4_F32 tiles.
// One block per plane. npad in {16, 32}; blockDim.x = 32 * (npad/16)^2.
// WMMA f32 16x16x4 register layouts (ISA 7.12.2):
//   A 16x4 : lane L -> row M=L%16; regs hold K = {0,1} (L<16) or {2,3}
//   B 4x16 : lane L -> col N=L%16; regs hold K = {0,1} (L<16) or {2,3}
//   C 16x16: lane L, reg g -> row M = g + (L<16 ? 0 : 8), col N = L%16
// ---------------------------------------------------------------------------
__global__ void blur_wmma_kernel(const float* __restrict__ src, float* __restrict__ dst,
                                 const float* __restrict__ Amat, int n, int npad) {
  __shared__ float Sin[32 * 32];
  __shared__ float ST[32 * 32];
  const int plane = blockIdx.x;
  const float* sp = src + (size_t)plane * n * n;
  float* dp = dst + (size_t)plane * n * n;
  const int tid = threadIdx.x;
  // load + zero-pad input tile
  for (int i = tid; i < npad * npad; i += blockDim.x) {
    int r = i / npad, c = i % npad;
    Sin[i] = (r < n && c < n) ? sp[r * n + c] : 0.f;
  }
  __syncthreads();

  const int wave = tid >> 5;
  const int lane = tid & 31;
  const int tcols = npad >> 4;
  const int ti = wave / tcols, tj = wave % tcols;
  const int M = lane & 15;                 // also the N index for B/C operands
  const int khalf = (lane < 16) ? 0 : 2;
  const int rowoff = (lane < 16) ? 0 : 8;

  // ---- T = In * A  (horizontal blur) ----
  v8f acc = {0.f, 0.f, 0.f, 0.f, 0.f, 0.f, 0.f, 0.f};
  for (int kb = 0; kb < npad; kb += 4) {
    v2f a, b;
    a.x = Sin[(ti * 16 + M) * npad + kb + khalf + 0];
    a.y = Sin[(ti * 16 + M) * npad + kb + khalf + 1];
    b.x = Amat[(kb + khalf + 0) * npad + tj * 16 + M];
    b.y = Amat[(kb + khalf + 1) * npad + tj * 16 + M];
    acc = __builtin_amdgcn_wmma_f32_16x16x4_f32(false, a, false, b, (short)0, acc,
                                                false, false);
  }
#pragma unroll
  for (int g = 0; g < 8; ++g)
    ST[(ti * 16 + g + rowoff) * npad + tj * 16 + M] = acc[g];
  __syncthreads();

  // ---- Out = A^T * T  (vertical blur) ----
  v8f oacc = {0.f, 0.f, 0.f, 0.f, 0.f, 0.f, 0.f, 0.f};
  for (int kb = 0; kb < npad; kb += 4) {
    v2f a, b;
    a.x = Amat[(kb + khalf + 0) * npad + ti * 16 + M];   // A^T[M][k] = A[k][M]
    a.y = Amat[(kb + khalf + 1) * npad + ti * 16 + M];
    b.x = ST[(kb + khalf + 0) * npad + tj * 16 + M];
    b.y = ST[(kb + khalf + 1) * npad + tj * 16 + M];
    oacc = __builtin_amdgcn_wmma_f32_16x16x4_f32(false, a, false, b, (short)0, oacc,
                                                 false, false);
  }
#pragma unroll
  for (int g = 0; g < 8; ++g) {
    int row = ti * 16 + g + rowoff;
    int col = tj * 16 + M;
    if (row < n && col < n) dp[row * n + col] = oacc[g];
  }
}

// ---------------------------------------------------------------------------
// cv2-style bilinear x2 upsample (half-pixel centers, clipped)
// ---------------------------------------------------------------------------
__device__ __forceinline__ void upw(int o, int n, int& c0, int& c1, float& t) {
  float c = (o + 0.5f) * 0.5f - 0.5f;
  c = fminf(fmaxf(c, 0.0f), (float)(n - 1));
  c0 = (int)c;
  c1 = min(c0 + 1, n - 1);
  t = c - (float)c0;
}

__global__ void upsample2x_kernel(const float* __restrict__ src, float* __restrict__ dst,
                                  int n) {
  const int no = n * 2;
  const int p = blockIdx.z;
  const int ox = blockIdx.x * 16 + threadIdx.x;
  const int oy = blockIdx.y * 16 + threadIdx.y;
  if (ox >= no || oy >= no) return;
  const float* sp = src + (size_t)p * n * n;
  int y0, y1, x0, x1;
  float ty, tx;
  upw(oy, n, y0, y1, ty);
  upw(ox, n, x0, x1, tx);
  float v0 = sp[y0 * n + x0] * (1.f - tx) + sp[y0 * n + x1] * tx;
  float v1 = sp[y1 * n + x0] * (1.f - tx) + sp[y1 * n + x1] * tx;
  dst[(size_t)p * no * no + oy * no + ox] = v0 * (1.f - ty) + v1 * ty;
}

// ---------------------------------------------------------------------------
// Final fused kernel: inline 512->1024 bilinear of the three sigma maps,
// MSR + color restoration + clip. One thread per (b,h,w); 3 channels together.
// ---------------------------------------------------------------------------
__global__ void msr_final_kernel(const float* __restrict__ x,
                                 const float* __restrict__ S30,
                                 const float* __restrict__ S150,
                                 const float* __restrict__ S300,
                                 float* __restrict__ out) {
  const int idx = blockIdx.x * blockDim.x + threadIdx.x;
  if (idx >= 8 * 1024 * 1024) return;
  const int b = idx >> 20;
  const int rem = idx & ((1 << 20) - 1);
  const int h = rem >> 10, w = rem & 1023;
  int y0, y1, x0, x1;
  float ty, tx;
  upw(h, 512, y0, y1, ty);
  upw(w, 512, x0, x1, tx);
  const float w00 = (1.f - ty) * (1.f - tx), w01 = (1.f - ty) * tx;
  const float w10 = ty * (1.f - tx), w11 = ty * tx;

  float img[3], lsum[3];
  float total = 0.f;
  const float* maps[3] = {S30, S150, S300};
#pragma unroll
  for (int c = 0; c < 3; ++c) {
    const int pl = b * 3 + c;
    float v = x[((size_t)pl << 20) + rem] * 255.0f + 1.0f;
    img[c] = v;
    total += v;
    const size_t pbase = (size_t)pl * 512 * 512;
    float ls = 0.f;
#pragma unroll
    for (int s = 0; s < 3; ++s) {
      const float* m = maps[s] + pbase;
      float F = w00 * m[y0 * 512 + x0] + w01 * m[y0 * 512 + x1] +
                w10 * m[y1 * 512 + x0] + w11 * m[y1 * 512 + x1];
      ls += log10f(F + EPSF);
    }
    lsum[c] = ls;
  }
#pragma unroll
  for (int c = 0; c < 3; ++c) {
    float retinex = log10f(img[c] + EPSF) - lsum[c] * (1.0f / 3.0f);
    float cr = log10f(img[c] * 2.0f / (total + EPSF) + 1.0f);
    float e = retinex * cr * 2700.0f + 128.0f;           // COLOR_GAIN*GAIN, OFFSET
    e = fminf(fmaxf(e, 0.f), 255.f) * (1.0f / 255.0f);
    out[((size_t)(b * 3 + c) << 20) + rem] = e;
  }
}

// ---------------------------------------------------------------------------
extern "C" void kernel_launch(void* const* d_in, const int* in_sizes, int n_in,
                              void* d_out, int out_size, void* d_ws, size_t ws_size,
                              hipStream_t stream) {
  (void)in_sizes; (void)n_in; (void)out_size; (void)ws_size;
  const float* x = (const float*)d_in[0];
  float* out = (float*)d_out;
  float* ws = (float*)d_ws;

  size_t off = 0;
  auto take = [&](size_t nf) { float* p = ws + off; off += nf; return p; };
  const size_t NP = NPLANES;
  float* A32 = take(32 * 32);
  float* A8  = take(16 * 16);
  float* A4  = take(16 * 16);
  float* P1 = take(NP * 512 * 512);
  float* P2 = take(NP * 256 * 256);
  float* P3 = take(NP * 128 * 128);
  float* P4 = take(NP * 64 * 64);
  float* P5 = take(NP * 32 * 32);
  float* P6 = take(NP * 16 * 16);
  float* P7 = take(NP * 8 * 8);
  float* P8 = take(NP * 4 * 4);
  float* B30  = take(NP * 32 * 32);
  float* B150 = take(NP * 8 * 8);
  float* B300 = take(NP * 4 * 4);
  float* S30  = take(NP * 512 * 512);
  float* S150 = take(NP * 512 * 512);
  float* S300 = take(NP * 512 * 512);
  float* T8   = take(NP * 8 * 8);
  float* T16  = take(NP * 16 * 16);
  float* T32  = take(NP * 32 * 32);
  float* T64  = take(NP * 64 * 64);
  float* T128 = take(NP * 128 * 128);
  float* T256 = take(NP * 256 * 256);

  // Blur matrices: sigma30 -> ksize7 @32; sigma150 -> ksize9 @8; sigma300 -> ksize9 @4
  build_blur_matrix_kernel<<<1, 1, 0, stream>>>(A32, 32, 32, 7);
  build_blur_matrix_kernel<<<1, 1, 0, stream>>>(A8, 8, 16, 9);
  build_blur_matrix_kernel<<<1, 1, 0, stream>>>(A4, 4, 16, 9);

  auto pyr = [&](const float* src, float* dst, int n, int fromInput) {
    int no = n >> 1;
    dim3 g((no + 15) / 16, (no + 15) / 16, (unsigned)NP), b(16, 16);
    pyrdown_kernel<<<g, b, 0, stream>>>(src, dst, n, fromInput);
  };
  pyr(x, P1, 1024, 1);
  pyr(P1, P2, 512, 0);
  pyr(P2, P3, 256, 0);
  pyr(P3, P4, 128, 0);
  pyr(P4, P5, 64, 0);
  pyr(P5, P6, 32, 0);
  pyr(P6, P7, 16, 0);
  pyr(P7, P8, 8, 0);

  // WMMA blurs at the recursion leaves
  blur_wmma_kernel<<<(unsigned)NP, 128, 0, stream>>>(P5, B30, A32, 32, 32);
  blur_wmma_kernel<<<(unsigned)NP, 32, 0, stream>>>(P7, B150, A8, 8, 16);
  blur_wmma_kernel<<<(unsigned)NP, 32, 0, stream>>>(P8, B300, A4, 4, 16);

  auto up = [&](const float* src, float* dst, int n) {
    int no = n * 2;
    dim3 g((no + 15) / 16, (no + 15) / 16, (unsigned)NP), b(16, 16);
    upsample2x_kernel<<<g, b, 0, stream>>>(src, dst, n);
  };
  // sigma=30: 32 -> 512
  up(B30, T64, 32); up(T64, T128, 64); up(T128, T256, 128); up(T256, S30, 256);
  // sigma=150: 8 -> 512
  up(B150, T16, 8); up(T16, T32, 16); up(T32, T64, 32);
  up(T64, T128, 64); up(T128, T256, 128); up(T256, S150, 256);
  // sigma=300: 4 -> 512
  up(B300, T8, 4); up(T8, T16, 8); up(T16, T32, 16); up(T32, T64, 32);
  up(T64, T128, 64); up(T128, T256, 128); up(T256, S300, 256);

  // Fused final: inline 512->1024 upsample + MSR enhancement
  const int npix = 8 * 1024 * 1024;
  msr_final_kernel<<<(npix + 255) / 256, 256, 0, stream>>>(x, S30, S150, S300, out);
}